// SparseEdgeEmbeddingV4_38878043964001
// MI455X (gfx1250) — compile-verified
//
#include <hip/hip_runtime.h>
#include <hip/hip_bf16.h>

// ---------------------------------------------------------------------------
// Problem constants (match the JAX reference)
// ---------------------------------------------------------------------------
#define N_PTS 8192
#define KNN   32
#define EDGES (N_PTS * KNN)   // 262144
#define NOUT  64

typedef float v2f __attribute__((ext_vector_type(2)));
typedef float v8f __attribute__((ext_vector_type(8)));

// Output layout (flat elements, return-order concatenation):
//   k        : (EDGES+1)*NOUT f32   @ 0
//   row_idx  : (EDGES+1)*KNN  i32   @ OFF_ROW
//   col_idx  : (EDGES+1)*KNN  i32   @ OFF_COL
//   all_ij_id: EDGES*2        i32   @ OFF_IJ
#define OFF_ROW ((EDGES + 1) * NOUT)
#define OFF_COL (OFF_ROW + (EDGES + 1) * KNN)
#define OFF_IJ  (OFF_COL + (EDGES + 1) * KNN)

// ---------------------------------------------------------------------------
// Kernel 0: zero a region of ints (counters / colmat must start at 0 each call)
// ---------------------------------------------------------------------------
__global__ void zero_kernel(int* __restrict__ p, int n) {
    int g = blockIdx.x * blockDim.x + threadIdx.x;
    if (g < n) p[g] = 0;
}

// ---------------------------------------------------------------------------
// Kernel 1: fused distance + top-32 selection using V_WMMA_F32_16X16X4_F32.
// One wave (32 lanes) owns a 16-query strip. Per candidate tile of 16 points:
//   G = A(16x4) x B(4x16) via WMMA, d2 = |q|^2 + |c|^2 - 2G, threshold-filtered
//   insertion into a sorted 32-entry LDS list per query row.
// f32 A-matrix 16x4 layout: lanes 0-15 hold K=0,1 in v0,v1; lanes 16-31 K=2,3.
// B (4x16) mirrors it. Each lane loads all 3 dims of its point unconditionally
// (branchless, always in-bounds) so norms need no cross-lane combine and the
// WMMA operand packing is a pair of v_cndmask selects.
// ---------------------------------------------------------------------------
__global__ __launch_bounds__(32) void knn_wmma_kernel(const float* __restrict__ coord,
                                                      float* __restrict__ dflat,
                                                      int*   __restrict__ colflat) {
    __shared__ float td[16 * 32];   // sorted ascending squared distances
    __shared__ int   ti[16 * 32];   // matching candidate indices
    __shared__ float qn_lds[16];    // |q|^2 for the 16 query rows

    const int lane  = threadIdx.x;
    const int m     = lane & 15;    // point index within tile
    const int half  = lane >> 4;    // which K-half this lane carries
    const int qbase = blockIdx.x * 16;

    // A tile (queries): load all 3 dims, pack K-half branchlessly, pad K=3 -> 0.
    {
        const int q = qbase + m;
        const float q0 = coord[q * 3 + 0];
        const float q1 = coord[q * 3 + 1];
        const float q2 = coord[q * 3 + 2];
        if (half == 0) qn_lds[m] = q0 * q0 + q1 * q1 + q2 * q2;
    }

    #pragma unroll
    for (int r = 0; r < 16; ++r) { td[r * 32 + lane] = 3.4e38f; ti[r * 32 + lane] = 0; }
    __syncthreads();

    // Re-read (registers) for the A operand.
    const int q = qbase + m;
    const float qa0 = coord[q * 3 + 0];
    const float qa1 = coord[q * 3 + 1];
    const float qa2 = coord[q * 3 + 2];
    v2f a;
    a.x = half ? qa2 : qa0;
    a.y = half ? 0.0f : qa1;

    // Per-lane row norms for the D-tile rows this lane owns (row = 8*half + j).
    float qn[8];
    #pragma unroll
    for (int j = 0; j < 8; ++j) qn[j] = qn_lds[half * 8 + j];

    float worst[8];
    #pragma unroll
    for (int j = 0; j < 8; ++j) worst[j] = 3.4e38f;

    for (int t = 0; t < N_PTS / 16; ++t) {
        const int cb = t * 16;
        if (t + 1 < N_PTS / 16) __builtin_prefetch(&coord[(cb + 16) * 3], 0, 1);

        const int   p  = cb + m;
        const float c0 = coord[p * 3 + 0];
        const float c1 = coord[p * 3 + 1];
        const float c2 = coord[p * 3 + 2];
        const float cn = c0 * c0 + c1 * c1 + c2 * c2;  // full |c_n|^2, n == lane&15
        v2f b;
        b.x = half ? c2 : c0;
        b.y = half ? 0.0f : c1;

        v8f acc = {0.f, 0.f, 0.f, 0.f, 0.f, 0.f, 0.f, 0.f};
        acc = __builtin_amdgcn_wmma_f32_16x16x4_f32(
            /*neg_a=*/false, a, /*neg_b=*/false, b,
            /*c_mod=*/(short)0, acc, /*reuse_a=*/false, /*reuse_b=*/false);

        const int cidx = cb + m;            // candidate id for this lane's column
        #pragma unroll
        for (int j = 0; j < 8; ++j) {
            const int   rowm = half * 8 + j;
            const float d2v  = fmaxf(qn[j] + cn - 2.0f * acc[j], 0.0f);
            unsigned long long mask = __ballot(d2v < worst[j]);
            while (mask) {                  // rare after warm-up (~k*ln(n/k) hits/row)
                const int src = (int)__ffsll((unsigned long long)mask) - 1;
                if (lane == src) {          // serialized sorted insertion
                    int k = 31;
                    while (k > 0 && td[rowm * 32 + k - 1] > d2v) {
                        td[rowm * 32 + k] = td[rowm * 32 + k - 1];
                        ti[rowm * 32 + k] = ti[rowm * 32 + k - 1];
                        --k;
                    }
                    td[rowm * 32 + k] = d2v;
                    ti[rowm * 32 + k] = cidx;
                }
                asm volatile("s_wait_dscnt 0" ::: "memory");   // cross-lane LDS RAW
                worst[j] = td[rowm * 32 + 31];
                mask &= ~(1ull << src);
                mask &= __ballot(d2v < worst[j]);
            }
        }
    }
    __syncthreads();

    #pragma unroll
    for (int r = 0; r < 16; ++r) {
        const int grow = qbase + r;
        dflat[grow * KNN + lane]   = td[r * 32 + lane];
        colflat[grow * KNN + lane] = ti[r * 32 + lane];
    }
}

// ---------------------------------------------------------------------------
// Kernel 2: per-edge pass — all_ij_id output + in-degree histogram.
// ---------------------------------------------------------------------------
__global__ void edge_kernel(const int* __restrict__ colflat,
                            int* __restrict__ counts,
                            int* __restrict__ out_ij) {
    int e = blockIdx.x * blockDim.x + threadIdx.x;
    if (e >= EDGES) return;
    const int c = colflat[e];
    out_ij[2 * e]     = e >> 5;   // source node
    out_ij[2 * e + 1] = c;        // target node
    atomicAdd(&counts[c], 1);
}

// ---------------------------------------------------------------------------
// Kernel 3: exclusive prefix scan of counts[8192] -> starts[8192] (1 block).
// ---------------------------------------------------------------------------
__global__ __launch_bounds__(256) void scan_kernel(const int* __restrict__ counts,
                                                   int* __restrict__ starts) {
    __shared__ int lds[256];
    const int t = threadIdx.x;
    const int base = t * 32;
    int sum = 0;
    for (int i = 0; i < 32; ++i) sum += counts[base + i];
    lds[t] = sum;
    __syncthreads();
    for (int off = 1; off < 256; off <<= 1) {
        int v = (t >= off) ? lds[t - off] : 0;
        __syncthreads();
        lds[t] += v;
        __syncthreads();
    }
    int run = lds[t] - sum;   // exclusive prefix of this chunk
    for (int i = 0; i < 32; ++i) { starts[base + i] = run; run += counts[base + i]; }
}

// ---------------------------------------------------------------------------
// Kernel 4: scatter edges into per-target groups (order-free; ranks fix order).
// ---------------------------------------------------------------------------
__global__ void scatter_kernel(const int* __restrict__ colflat,
                               const float* __restrict__ dflat,
                               const int* __restrict__ starts,
                               int* __restrict__ cursor,
                               int* __restrict__ ge, float* __restrict__ gd) {
    int e = blockIdx.x * blockDim.x + threadIdx.x;
    if (e >= EDGES) return;
    const int c   = colflat[e];
    const int pos = starts[c] + atomicAdd(&cursor[c], 1);
    ge[pos] = e;
    gd[pos] = dflat[e];
}

// ---------------------------------------------------------------------------
// Kernel 5: per-target-node ranking -> colmat (1-based edge ids, 0-padded).
//   count <= 32 : rank by edge id (stable edge-id order, zero-pad at end)
//   count  > 32 : rank by (distance, edge id); rank >= 32 dropped
// Deterministic regardless of the atomic scatter order.
// ---------------------------------------------------------------------------
__global__ __launch_bounds__(32) void rank_kernel(const int* __restrict__ counts,
                                                  const int* __restrict__ starts,
                                                  const int* __restrict__ ge,
                                                  const float* __restrict__ gd,
                                                  int* __restrict__ colmat) {
    __shared__ int   se[1024];
    __shared__ float sd[1024];
    const int t    = blockIdx.x;
    const int lane = threadIdx.x;
    int cnt = counts[t];
    if (cnt > 1024) cnt = 1024;   // safety clamp (far above any realistic in-degree)
    const int s0 = starts[t];
    for (int i = lane; i < cnt; i += 32) { se[i] = ge[s0 + i]; sd[i] = gd[s0 + i]; }
    __syncthreads();

    const bool mode = cnt > KNN;
    for (int i = lane; i < cnt; i += 32) {
        const int   ei = se[i];
        const float di = sd[i];
        int rank = 0;
        for (int j = 0; j < cnt; ++j) {
            const int   ej = se[j];
            const float dj = sd[j];
            const bool less = mode ? ((dj < di) || (dj == di && ej < ei)) : (ej < ei);
            rank += less ? 1 : 0;
        }
        if (rank < KNN) colmat[t * KNN + rank] = ei + 1;   // colmat pre-zeroed
    }
}

// ---------------------------------------------------------------------------
// Kernel 6: row_idx expansion — pure function of the index.
// ---------------------------------------------------------------------------
__global__ void rowidx_kernel(int* __restrict__ out_row) {
    int g = blockIdx.x * blockDim.x + threadIdx.x;
    if (g >= (EDGES + 1) * KNN) return;
    const int r = g >> 5, j = g & 31;
    out_row[g] = (r == 0) ? 0 : ((((r - 1) >> 5) << 5) + j);
}

// ---------------------------------------------------------------------------
// Kernel 7: col_idx expansion — broadcast colmat row of the SOURCE node.
// ---------------------------------------------------------------------------
__global__ void colidx_kernel(const int* __restrict__ colmat, int* __restrict__ out_col) {
    int g = blockIdx.x * blockDim.x + threadIdx.x;
    if (g >= (EDGES + 1) * KNN) return;
    const int r = g >> 5, j = g & 31;
    out_col[g] = (r == 0) ? 0 : colmat[((r - 1) >> 5) * KNN + j];
}

// ---------------------------------------------------------------------------
// Kernel 8: Gaussian kernel embedding k[e+1][s] = exp(-d2 / (2*sigma_s^2)).
// ---------------------------------------------------------------------------
__global__ void kexp_kernel(const float* __restrict__ dflat, float* __restrict__ kout) {
    int g = blockIdx.x * blockDim.x + threadIdx.x;
    if (g >= (EDGES + 1) * NOUT) return;
    const int e1 = g >> 6, s = g & 63;
    float val = 0.0f;
    if (e1 > 0) {
        const float d   = dflat[e1 - 1];
        const float sig = 0.1f + (float)s * (4.9f / 63.0f);
        const float inv = 1.0f / (2.0f * sig * sig);
        val = __expf(-d * inv);
    }
    kout[g] = val;
}

// ---------------------------------------------------------------------------
// Launch wiring. Workspace layout (4-byte units):
//   dflat[E] | colflat[E] | counts[N] | cursor[N] | starts[N]
//   | ge[E] | gd[E] | colmat[N*KNN]          (total ~5.1 MB)
// ---------------------------------------------------------------------------
extern "C" void kernel_launch(void* const* d_in, const int* in_sizes, int n_in,
                              void* d_out, int out_size, void* d_ws, size_t ws_size,
                              hipStream_t stream) {
    const float* coord = (const float*)d_in[0];

    float* kout   = (float*)d_out;
    int*   rowout = (int*)d_out + OFF_ROW;
    int*   colout = (int*)d_out + OFF_COL;
    int*   ijout  = (int*)d_out + OFF_IJ;

    int*   base    = (int*)d_ws;
    float* dflat   = (float*)d_ws;
    int*   colflat = base + EDGES;
    int*   counts  = base + 2 * EDGES;
    int*   cursor  = counts + N_PTS;
    int*   starts  = cursor + N_PTS;
    int*   ge      = starts + N_PTS;
    float* gd      = (float*)(ge + EDGES);
    int*   colmat  = (int*)(gd + EDGES);

    // counts + cursor are contiguous -> one zero pass; colmat separately.
    zero_kernel<<<(2 * N_PTS + 255) / 256, 256, 0, stream>>>(counts, 2 * N_PTS);
    zero_kernel<<<(N_PTS * KNN + 255) / 256, 256, 0, stream>>>(colmat, N_PTS * KNN);

    knn_wmma_kernel<<<N_PTS / 16, 32, 0, stream>>>(coord, dflat, colflat);

    edge_kernel<<<(EDGES + 255) / 256, 256, 0, stream>>>(colflat, counts, ijout);
    scan_kernel<<<1, 256, 0, stream>>>(counts, starts);
    scatter_kernel<<<(EDGES + 255) / 256, 256, 0, stream>>>(colflat, dflat, starts,
                                                            cursor, ge, gd);
    rank_kernel<<<N_PTS, 32, 0, stream>>>(counts, starts, ge, gd, colmat);

    rowidx_kernel<<<((EDGES + 1) * KNN + 255) / 256, 256, 0, stream>>>(rowout);
    colidx_kernel<<<((EDGES + 1) * KNN + 255) / 256, 256, 0, stream>>>(colmat, colout);
    kexp_kernel<<<((EDGES + 1) * NOUT + 255) / 256, 256, 0, stream>>>(dflat, kout);
}